// BPGNN_37709812859111
// MI455X (gfx1250) — compile-verified
//
#include <hip/hip_runtime.h>
#include <hip/hip_bf16.h>

// ---------------------------------------------------------------------------
// BP-GNN on MI455X (gfx1250, wave32).
// Dense J is never streamed: all reductions touch only the 65536 nonzeros,
// enumerated by the sorted edge lists. Node MLP/GAT GEMMs use fp32 WMMA
// (V_WMMA_F32_16X16X4_F32) so numerics match the fp32 reference.
// ---------------------------------------------------------------------------

typedef __attribute__((ext_vector_type(2))) float v2f;
typedef __attribute__((ext_vector_type(8))) float v8f;

#define HIDN 64
#define NSTEP 10   // reference constant N_STEP (device scalar; must be host-known)

__device__ __forceinline__ int lower_bound_i(const int* a, int n, int key) {
    int lo = 0, hi = n;
    while (lo < hi) {
        int mid = (lo + hi) >> 1;
        if (a[mid] < key) lo = mid + 1; else hi = mid;
    }
    return lo;
}

__global__ void k_zero(float* p, int n) {
    int i = blockIdx.x * blockDim.x + threadIdx.x;
    if (i < n) p[i] = 0.0f;
}

// Jv[e] = J[row[e], col[e]]  (only touch of the dense 64MB matrix: 65536 gathers)
__global__ void k_edge_gather(const float* __restrict__ J, const int* __restrict__ row,
                              const int* __restrict__ col, float* __restrict__ Jv,
                              int E, int N) {
    int e = blockIdx.x * blockDim.x + threadIdx.x;
    if (e >= E) return;
    Jv[e] = J[(size_t)row[e] * (size_t)N + (size_t)col[e]];
}

// Per-node: edge segment [start,end) via binary search on sorted `row`,
// degree = count, rowsum = sum of Jv over segment; feat = [-b, b, deg, rowsum]
__global__ void k_node_feat(const int* __restrict__ row, const float* __restrict__ Jv,
                            const float* __restrict__ bias,
                            int* __restrict__ nodeStart, int* __restrict__ nodeCnt,
                            float* __restrict__ feat, int N, int E) {
    int i = blockIdx.x * blockDim.x + threadIdx.x;
    if (i >= N) return;
    int s = lower_bound_i(row, E, i);
    int t = lower_bound_i(row, E, i + 1);
    nodeStart[i] = s;
    nodeCnt[i] = t - s;
    float rs = 0.0f;
    for (int e = s; e < t; ++e) rs += Jv[e];
    float b = bias[i];
    feat[4 * i + 0] = -b;
    feat[4 * i + 1] = b;
    feat[4 * i + 2] = (float)(t - s);
    feat[4 * i + 3] = rs;
}

// Node networks via fp32 WMMA. Block = 128 threads = 4 wave32; 16-node tile.
// Wave w computes output columns [16w, 16w+16).
//   GAT:  Wh = feat @ gat_W            (K=4, one WMMA)
//   MLP:  h1 = relu(feat@W1+b1)        (K=4, one WMMA)
//         h2 = relu(h1@W2+b2)          (K=64, 16 chained WMMA)
//         Ci = h2@W3 + b3
//   f1 = Wh@a[:64], f2 = Wh@a[64:]
__global__ __launch_bounds__(128) void k_node_net(
    const float* __restrict__ feat,
    const float* __restrict__ gat_W, const float* __restrict__ gat_a,
    const float* __restrict__ W1, const float* __restrict__ b1,
    const float* __restrict__ W2, const float* __restrict__ b2,
    const float* __restrict__ W3, const float* __restrict__ b3,
    float* __restrict__ f1, float* __restrict__ f2,
    float* __restrict__ Ci_ws, float* __restrict__ Ci_out, int N)
{
    __shared__ float ldsA[16 * 68];   // Wh tile, later reused for h2 tile
    __shared__ float ldsH1[16 * 68];  // h1 tile (A operand of GEMM2)

    const int tid  = threadIdx.x;
    const int wave = tid >> 5;
    const int lane = tid & 31;
    const int mrow = lane & 15;      // A-matrix M index / B-matrix N index
    const int hi   = lane >> 4;      // lane half selects K pair (A/B) and M offset (C/D)
    const int kh   = hi * 2;
    const int ncol = (wave << 4) + mrow;     // output column owned by this lane
    const int mbase = blockIdx.x * 16;

    // A fragment: feat[mbase+mrow][kh], [kh+1]   (16x4 fp32 A layout)
    v2f a;
    a.x = feat[(mbase + mrow) * 4 + kh];
    a.y = feat[(mbase + mrow) * 4 + kh + 1];
    // B fragments: W[kh][ncol], W[kh+1][ncol]    (4x16 fp32 B layout)
    v2f bg, bw1;
    bg.x  = gat_W[kh * HIDN + ncol];  bg.y  = gat_W[(kh + 1) * HIDN + ncol];
    bw1.x = W1[kh * HIDN + ncol];     bw1.y = W1[(kh + 1) * HIDN + ncol];

    v8f cz = {0.f, 0.f, 0.f, 0.f, 0.f, 0.f, 0.f, 0.f};
    v8f cWh = __builtin_amdgcn_wmma_f32_16x16x4_f32(false, a, false, bg,  (short)0, cz, false, false);
    v8f cH1 = __builtin_amdgcn_wmma_f32_16x16x4_f32(false, a, false, bw1, (short)0, cz, false, false);

    float bb1 = b1[ncol];
#pragma unroll
    for (int v = 0; v < 8; ++v) {
        int r = v + 8 * hi;                 // C/D layout: M = v + 8*(lane>=16)
        ldsA[r * 68 + ncol] = cWh[v];
        float h = cH1[v] + bb1;
        ldsH1[r * 68 + ncol] = h > 0.f ? h : 0.f;
    }
    __syncthreads();

    // GEMM2: h1[16x64] @ W2[64x64], chained over 16 K=4 WMMAs (uniform, EXEC all-1)
    v8f acc = cz;
#pragma unroll
    for (int kk = 0; kk < 16; ++kk) {
        v2f a2, b2f;
        a2.x  = ldsH1[mrow * 68 + kk * 4 + kh];
        a2.y  = ldsH1[mrow * 68 + kk * 4 + kh + 1];
        b2f.x = W2[(kk * 4 + kh) * HIDN + ncol];
        b2f.y = W2[(kk * 4 + kh + 1) * HIDN + ncol];
        acc = __builtin_amdgcn_wmma_f32_16x16x4_f32(false, a2, false, b2f, (short)0, acc, false, false);
    }

    // attention scalars from Wh tile
    if (tid < 16) {
        float s1 = 0.f, s2 = 0.f;
        for (int n = 0; n < HIDN; ++n) {
            float w = ldsA[tid * 68 + n];
            s1 += w * gat_a[n];
            s2 += w * gat_a[HIDN + n];
        }
        f1[mbase + tid] = s1;
        f2[mbase + tid] = s2;
    }
    __syncthreads();   // readers of ldsA done before overwrite

    float bb2 = b2[ncol];
#pragma unroll
    for (int v = 0; v < 8; ++v) {
        int r = v + 8 * hi;
        float h = acc[v] + bb2;
        ldsA[r * 68 + ncol] = h > 0.f ? h : 0.f;   // h2
    }
    __syncthreads();

    if (tid < 16) {
        float s = 0.f;
        for (int k = 0; k < HIDN; ++k) s += ldsA[tid * 68 + k] * W3[k];
        s += b3[0];
        Ci_ws[mbase + tid]  = s;
        Ci_out[mbase + tid] = s;
    }
}

// Cij_e[e] = 0.5*(exp(lrelu(f1[r]+f2[c])) + exp(lrelu(f1[c]+f2[r])))
__global__ void k_edge_cij(const int* __restrict__ row, const int* __restrict__ col,
                           const float* __restrict__ f1, const float* __restrict__ f2,
                           float* __restrict__ Cij, int E) {
    int e = blockIdx.x * blockDim.x + threadIdx.x;
    if (e >= E) return;
    int r = row[e], c = col[e];
    float e1 = f1[r] + f2[c]; e1 = e1 > 0.f ? e1 : 0.2f * e1;
    float e2 = f1[c] + f2[r]; e2 = e2 > 0.f ? e2 : 0.2f * e2;
    Cij[e] = 0.5f * (expf(e1) + expf(e2));
}

// dom = signclamp(Ci + sum_j Cij, 0.1); bias_scale = bias/dom
__global__ void k_node_dom(const int* __restrict__ nodeStart, const int* __restrict__ nodeCnt,
                           const float* __restrict__ Cij, const float* __restrict__ Ci_ws,
                           const float* __restrict__ bias,
                           float* __restrict__ dom_ws, float* __restrict__ dom_out,
                           float* __restrict__ bs, int N) {
    int i = blockIdx.x * blockDim.x + threadIdx.x;
    if (i >= N) return;
    int s = nodeStart[i], c = nodeCnt[i];
    float sc = 0.0f;
    for (int e = s; e < s + c; ++e) sc += Cij[e];
    float d0 = Ci_ws[i] + sc;
    float sgn = (d0 > 0.f) ? 1.f : ((d0 < 0.f) ? -1.f : 0.f);   // jnp.sign semantics
    float d = sgn * fmaxf(fabsf(d0), 0.1f);
    dom_ws[i]  = d;
    dom_out[i] = d;
    bs[i] = bias[i] / d;
}

// node_update: nm[j] = (sum over j's out-segment of em[rev[e]]) / dom[j]
// (in-edges of j are exactly rev[] of its out-edges -> deterministic, atomic-free)
__global__ void k_node_update(const float* __restrict__ em,
                              const int* __restrict__ nodeStart, const int* __restrict__ nodeCnt,
                              const int* __restrict__ rev, const float* __restrict__ dom,
                              const float* __restrict__ bs,
                              float* __restrict__ hidden, float* __restrict__ lnZ, int N) {
    int i = blockIdx.x * blockDim.x + threadIdx.x;
    if (i >= N) return;
    int s = nodeStart[i], c = nodeCnt[i];
    float nm0 = 0.f, nm1 = 0.f;
    for (int e = s; e < s + c; ++e) {
        int rv = rev[e];
        nm0 += em[2 * rv];
        nm1 += em[2 * rv + 1];
    }
    float d = dom[i];
    nm0 /= d; nm1 /= d;
    float b = bs[i];
    float t0 = -b + nm0, t1 = b + nm1;
    float m = fmaxf(t0, t1);
    float lz = m + logf(expf(t0 - m) + expf(t1 - m));
    hidden[2 * i]     = t0 - lz;
    hidden[2 * i + 1] = t1 - lz;
    lnZ[i] = lz;
}

// BP edge update, 2x2 collapsed analytically:
//   L[y] = max_x(Jxy[x][y] + hidden[row][x] + lnZ[row] - em[rev][x])
//   em_new[y] = C*(L[y] - logsumexp(L0,L1))
__global__ void k_bp_edge(const int* __restrict__ row, const int* __restrict__ rev,
                          const float* __restrict__ Cij, const float* __restrict__ Jv,
                          const float* __restrict__ hidden, const float* __restrict__ lnZ,
                          const float* __restrict__ em, float* __restrict__ emn, int E) {
    int e = blockIdx.x * blockDim.x + threadIdx.x;
    if (e >= E) return;
    int r = row[e], rv = rev[e];
    float C = Cij[e];
    float Jx = Jv[e] / C;
    float lz = lnZ[r];
    float g0 = hidden[2 * r]     + lz - em[2 * rv];
    float g1 = hidden[2 * r + 1] + lz - em[2 * rv + 1];
    float L0 = fmaxf( Jx + g0, -Jx + g1);
    float L1 = fmaxf(-Jx + g0,  Jx + g1);
    float m  = fmaxf(L0, L1);
    float ls = m + logf(expf(L0 - m) + expf(L1 - m));
    emn[2 * e]     = C * (L0 - ls);
    emn[2 * e + 1] = C * (L1 - ls);
}

// readout = exp(hidden); per-block partial of Ci . node_H
__global__ void k_final_node(const float* __restrict__ hidden, const float* __restrict__ Ci_ws,
                             float* __restrict__ out_read, float* __restrict__ partN, int N) {
    __shared__ float red[256];
    int i = blockIdx.x * blockDim.x + threadIdx.x;
    float val = 0.f;
    if (i < N) {
        float r0 = expf(hidden[2 * i]);
        float r1 = expf(hidden[2 * i + 1]);
        out_read[2 * i]     = r0;
        out_read[2 * i + 1] = r1;
        float nH = -(r0 * logf(r0 + 1e-16f) + r1 * logf(r1 + 1e-16f));
        val = Ci_ws[i] * nH;
    }
    red[threadIdx.x] = val;
    __syncthreads();
    for (int s = 128; s > 0; s >>= 1) {
        if (threadIdx.x < s) red[threadIdx.x] += red[threadIdx.x + s];
        __syncthreads();
    }
    if (threadIdx.x == 0) partN[blockIdx.x] = red[0];
}

// pairwise readout per upper edge + per-block partial of Cij_u . edge_H
__global__ void k_final_edge(const int* __restrict__ u2e, const int* __restrict__ rev,
                             const int* __restrict__ ru, const int* __restrict__ cu,
                             const float* __restrict__ Cij, const float* __restrict__ Jv,
                             const float* __restrict__ hidden, const float* __restrict__ em,
                             float* __restrict__ out_pw, float* __restrict__ out_Cij,
                             float* __restrict__ partE, int Eu) {
    __shared__ float red[256];
    int u = blockIdx.x * blockDim.x + threadIdx.x;
    float val = 0.f;
    if (u < Eu) {
        int e  = u2e[u];
        int er = rev[e];
        int ic = cu[u];     // temp_i from readout[cu]
        int jr = ru[u];     // temp_j from readout[ru]
        float C  = Cij[e];
        out_Cij[u] = C;
        float Js = Jv[e] / C;
        float ti0 = expf(hidden[2 * ic]     - em[2 * e]);
        float ti1 = expf(hidden[2 * ic + 1] - em[2 * e + 1]);
        float tj0 = expf(hidden[2 * jr]     - em[2 * er]);
        float tj1 = expf(hidden[2 * jr + 1] - em[2 * er + 1]);
        float eJp = expf(Js), eJm = expf(-Js);
        float p00 = eJp * ti0 * tj0;   // pw[x=0][y=0]
        float p01 = eJm * ti1 * tj0;   // pw[0][1]
        float p10 = eJm * ti0 * tj1;   // pw[1][0]
        float p11 = eJp * ti1 * tj1;   // pw[1][1]
        float inv = 1.f / (p00 + p01 + p10 + p11);
        p00 *= inv; p01 *= inv; p10 *= inv; p11 *= inv;
        out_pw[4 * u]     = p00;
        out_pw[4 * u + 1] = p01;
        out_pw[4 * u + 2] = p10;
        out_pw[4 * u + 3] = p11;
        float eH = -(p00 * logf(p00 + 1e-16f) + p01 * logf(p01 + 1e-16f) +
                     p10 * logf(p10 + 1e-16f) + p11 * logf(p11 + 1e-16f));
        val = C * eH;
    }
    red[threadIdx.x] = val;
    __syncthreads();
    for (int s = 128; s > 0; s >>= 1) {
        if (threadIdx.x < s) red[threadIdx.x] += red[threadIdx.x + s];
        __syncthreads();
    }
    if (threadIdx.x == 0) partE[blockIdx.x] = red[0];
}

__global__ void k_reduce(const float* __restrict__ partN, int nN,
                         const float* __restrict__ partE, int nE,
                         float* __restrict__ out_tot, float* __restrict__ out_ne,
                         float* __restrict__ out_ee) {
    __shared__ float red[256];
    float a = 0.f;
    for (int i = threadIdx.x; i < nN; i += 256) a += partN[i];
    red[threadIdx.x] = a;
    __syncthreads();
    for (int s = 128; s > 0; s >>= 1) {
        if (threadIdx.x < s) red[threadIdx.x] += red[threadIdx.x + s];
        __syncthreads();
    }
    float ne = red[0];
    __syncthreads();
    a = 0.f;
    for (int i = threadIdx.x; i < nE; i += 256) a += partE[i];
    red[threadIdx.x] = a;
    __syncthreads();
    for (int s = 128; s > 0; s >>= 1) {
        if (threadIdx.x < s) red[threadIdx.x] += red[threadIdx.x + s];
        __syncthreads();
    }
    if (threadIdx.x == 0) {
        float ee = red[0];
        *out_ne  = ne;
        *out_ee  = ee;
        *out_tot = ne + ee;
    }
}

extern "C" void kernel_launch(void* const* d_in, const int* in_sizes, int n_in,
                              void* d_out, int out_size, void* d_ws, size_t ws_size,
                              hipStream_t stream) {
    const float* J     = (const float*)d_in[0];
    const float* bias  = (const float*)d_in[1];
    const float* gat_W = (const float*)d_in[2];
    const float* gat_a = (const float*)d_in[3];
    const float* W1    = (const float*)d_in[4];
    const float* b1    = (const float*)d_in[5];
    const float* W2    = (const float*)d_in[6];
    const float* b2    = (const float*)d_in[7];
    const float* W3    = (const float*)d_in[8];
    const float* b3    = (const float*)d_in[9];
    const int* row = (const int*)d_in[10];
    const int* col = (const int*)d_in[11];
    const int* rev = (const int*)d_in[12];
    const int* ru  = (const int*)d_in[13];
    const int* cu  = (const int*)d_in[14];
    const int* u2e = (const int*)d_in[15];
    // d_in[16] = N_step (10), d_in[17] = MPNNtype (unused): device scalars;
    // the BP iteration count must be host-known for graph capture -> NSTEP.

    const int N  = in_sizes[1];
    const int E  = in_sizes[10];
    const int Eu = in_sizes[13];

    // ---- workspace layout (floats) ----
    float* ws = (float*)d_ws;
    size_t o = 0;
    float* Jv     = ws + o; o += (size_t)E;
    float* Cij    = ws + o; o += (size_t)E;
    float* feat   = ws + o; o += (size_t)N * 4;
    float* f1     = ws + o; o += (size_t)N;
    float* f2     = ws + o; o += (size_t)N;
    float* Ci_ws  = ws + o; o += (size_t)N;
    float* dom_ws = ws + o; o += (size_t)N;
    float* bs     = ws + o; o += (size_t)N;
    float* hidden = ws + o; o += (size_t)N * 2;
    float* lnZ    = ws + o; o += (size_t)N;
    float* emA    = ws + o; o += (size_t)E * 2;
    float* emB    = ws + o; o += (size_t)E * 2;
    int* nodeStart = (int*)(ws + o); o += (size_t)N;
    int* nodeCnt   = (int*)(ws + o); o += (size_t)N;
    float* partN  = ws + o; o += 256;
    float* partE  = ws + o; o += 256;

    // ---- output layout (flat tuple concat) ----
    float* out      = (float*)d_out;
    float* out_read = out;                                 // [N,2]
    float* out_pw   = out + (size_t)2 * N;                 // [Eu,4]
    float* out_tot  = out + (size_t)2 * N + (size_t)4 * Eu; // scalar
    float* out_ne   = out_tot + 1;                          // scalar
    float* out_ee   = out_tot + 2;                          // scalar
    float* out_Ci   = out_tot + 3;                          // [N]
    float* out_Cij  = out_Ci + N;                           // [Eu]
    float* out_dom  = out_Cij + Eu;                         // [N,1]

    const int B   = 256;
    const int gE  = (E + B - 1) / B;
    const int gN  = (N + B - 1) / B;
    const int gEu = (Eu + B - 1) / B;

    k_edge_gather<<<gE, B, 0, stream>>>(J, row, col, Jv, E, N);
    k_node_feat<<<gN, B, 0, stream>>>(row, Jv, bias, nodeStart, nodeCnt, feat, N, E);
    k_node_net<<<N / 16, 128, 0, stream>>>(feat, gat_W, gat_a, W1, b1, W2, b2, W3, b3,
                                           f1, f2, Ci_ws, out_Ci, N);
    k_edge_cij<<<gE, B, 0, stream>>>(row, col, f1, f2, Cij, E);
    k_node_dom<<<gN, B, 0, stream>>>(nodeStart, nodeCnt, Cij, Ci_ws, bias,
                                     dom_ws, out_dom, bs, N);

    k_zero<<<(2 * E + B - 1) / B, B, 0, stream>>>(emA, 2 * E);
    k_node_update<<<gN, B, 0, stream>>>(emA, nodeStart, nodeCnt, rev, dom_ws, bs,
                                        hidden, lnZ, N);

    float* cur = emA;
    float* nxt = emB;
    for (int step = 0; step < NSTEP; ++step) {
        k_bp_edge<<<gE, B, 0, stream>>>(row, rev, Cij, Jv, hidden, lnZ, cur, nxt, E);
        k_node_update<<<gN, B, 0, stream>>>(nxt, nodeStart, nodeCnt, rev, dom_ws, bs,
                                            hidden, lnZ, N);
        float* t = cur; cur = nxt; nxt = t;
    }

    k_final_node<<<gN, B, 0, stream>>>(hidden, Ci_ws, out_read, partN, N);
    k_final_edge<<<gEu, B, 0, stream>>>(u2e, rev, ru, cu, Cij, Jv, hidden, cur,
                                        out_pw, out_Cij, partE, Eu);
    k_reduce<<<1, B, 0, stream>>>(partN, gN, partE, gEu, out_tot, out_ne, out_ee);
}